// LegislativeGraphModel_44994077392975
// MI455X (gfx1250) — compile-verified
//
#include <hip/hip_runtime.h>
#include <math.h>

// ---------------------------------------------------------------------------
// LegislativeGraphModel on MI455X (gfx1250): fp32 WMMA 16x16x4 for all GEMMs,
// wave-per-edge message passing with atomics, fused LN+FFN per 16-row tile.
// ---------------------------------------------------------------------------

typedef __attribute__((ext_vector_type(2))) float v2f;
typedef __attribute__((ext_vector_type(8))) float v8f;

#define DIM   192
#define HID   768
#define LNPAD 200   // 16x192 tile padded row stride (floats)
#define UPAD  776   // 16x768 tile padded row stride (floats)

// ---------------------------------------------------------------------------
// Y[N,192] = X[N,192] @ W[192,192]   (N multiple of 16)
// block = 384 threads (12 waves); each wave owns one 16-wide column tile.
// X tile staged in LDS; W streamed from global (L2 resident, 147 KB).
// ---------------------------------------------------------------------------
__global__ __launch_bounds__(384) void gemm_x_w(
    const float* __restrict__ X, const float* __restrict__ W,
    float* __restrict__ Y, int nrows)
{
    __shared__ float ldsX[16 * LNPAD];

    const int m0 = blockIdx.x * 16;
    const int t  = threadIdx.x;

    // cooperative 16x192 tile load (float4 granularity)
    for (int i = t; i < 16 * 48; i += 384) {
        const int r = i / 48, c4 = i % 48;
        const float4 v = ((const float4*)(X + (size_t)(m0 + r) * DIM))[c4];
        float* p = &ldsX[r * LNPAD + c4 * 4];
        p[0] = v.x; p[1] = v.y; p[2] = v.z; p[3] = v.w;
    }
    __syncthreads();

    const int wave  = t >> 5;
    const int lane  = t & 31;
    const int laneM = lane & 15;      // A row / B,C column within tile
    const int half  = lane >> 4;
    const int kOff  = half * 2;
    const int n     = wave * 16 + laneM;

    v8f acc = {0.f,0.f,0.f,0.f,0.f,0.f,0.f,0.f};
    for (int k0 = 0; k0 < DIM; k0 += 4) {
        const int k = k0 + kOff;
        const v2f a = *(const v2f*)&ldsX[laneM * LNPAD + k];
        v2f b;
        b.x = W[(size_t)k * DIM + n];
        b.y = W[(size_t)(k + 1) * DIM + n];
        acc = __builtin_amdgcn_wmma_f32_16x16x4_f32(false, a, false, b,
                                                    (short)0, acc, false, false);
    }
    float* Yt = Y + (size_t)m0 * DIM + n;
#pragma unroll
    for (int r = 0; r < 8; ++r) {
        const int row = r + half * 8;
        Yt[(size_t)row * DIM] = acc[r];
    }
}

// ---------------------------------------------------------------------------
// One wave per edge.  lane owns 6 contiguous floats; 8 lanes == 1 head (48).
// l_h = dot(q_h, k_h + R_h)/sqrt(48) + ew ; softmax over the 4 heads;
// atomically accumulate p_h * v_h into out[dst].
// ---------------------------------------------------------------------------
__global__ __launch_bounds__(256) void edge_msg(
    const float* __restrict__ Q, const float* __restrict__ K,
    const float* __restrict__ V,
    const int* __restrict__ src, const int* __restrict__ dst,
    const float* __restrict__ ew, const float* __restrict__ R,
    float* __restrict__ out, int E)
{
    const int lane = threadIdx.x & 31;
    const int e    = blockIdx.x * 8 + (threadIdx.x >> 5);
    if (e >= E) return;

    const int s  = src[e];
    const int d  = dst[e];
    const int j0 = lane * 6;

    const float* Qp = Q + (size_t)d * DIM + j0;
    const float* Kp = K + (size_t)s * DIM + j0;
    const float* Vp = V + (size_t)s * DIM + j0;

    float vv[6];
    float part = 0.f;
#pragma unroll
    for (int i = 0; i < 6; ++i) {
        vv[i] = Vp[i];
        part  = fmaf(Qp[i], Kp[i] + R[j0 + i], part);
    }
    // reduce across the 8 lanes of this head
    part += __shfl_xor(part, 1, 32);
    part += __shfl_xor(part, 2, 32);
    part += __shfl_xor(part, 4, 32);

    const float l  = part * 0.14433756729740643f + ew[e];  // 1/sqrt(48)
    const float l0 = __shfl(l, 0, 32);
    const float l1 = __shfl(l, 8, 32);
    const float l2 = __shfl(l, 16, 32);
    const float l3 = __shfl(l, 24, 32);
    const float mx = fmaxf(fmaxf(l0, l1), fmaxf(l2, l3));
    const float den = expf(l0 - mx) + expf(l1 - mx) + expf(l2 - mx) + expf(l3 - mx);
    const float p  = expf(l - mx) / den;

    float* op = out + (size_t)d * DIM + j0;
#pragma unroll
    for (int i = 0; i < 6; ++i)
        atomicAdd(&op[i], p * vv[i]);
}

// ---------------------------------------------------------------------------
// Fused per-16-row-tile:  h_res = h + msg ; x = LN(h_res) ;
// u = gelu(x @ W1) (kept in LDS) ; out = h_res + u @ W2.
// block = 512 threads (16 waves).  LDS: 16x200 + 16x776 floats = 62.4 KB.
// ---------------------------------------------------------------------------
__global__ __launch_bounds__(512) void ffn_fused(
    const float* __restrict__ h,  const float* __restrict__ msg,
    const float* __restrict__ g,  const float* __restrict__ bb,
    const float* __restrict__ W1, const float* __restrict__ W2,
    float* __restrict__ out, int nrows)
{
    __shared__ float lnX[16 * LNPAD];
    __shared__ float U[16 * UPAD];

    const int t    = threadIdx.x;
    const int wave = t >> 5;
    const int lane = t & 31;
    const int m0   = blockIdx.x * 16;

    // ---- LayerNorm: wave w owns row w, lane owns 6 contiguous cols ----
    {
        const int row = wave;
        const size_t base = (size_t)(m0 + row) * DIM;
        float x[6];
        float s = 0.f;
#pragma unroll
        for (int i = 0; i < 6; ++i) {
            const int c = lane * 6 + i;
            x[i] = h[base + c] + msg[base + c];
            s += x[i];
        }
#pragma unroll
        for (int off = 1; off < 32; off <<= 1) s += __shfl_xor(s, off, 32);
        const float mu = s * (1.f / 192.f);
        float vs = 0.f;
#pragma unroll
        for (int i = 0; i < 6; ++i) { const float dx = x[i] - mu; vs += dx * dx; }
#pragma unroll
        for (int off = 1; off < 32; off <<= 1) vs += __shfl_xor(vs, off, 32);
        const float rstd = rsqrtf(vs * (1.f / 192.f) + 1e-5f);
#pragma unroll
        for (int i = 0; i < 6; ++i) {
            const int c = lane * 6 + i;
            lnX[row * LNPAD + c] = (x[i] - mu) * rstd * g[c] + bb[c];
        }
    }
    __syncthreads();

    const int laneM = lane & 15;
    const int half  = lane >> 4;
    const int kOff  = half * 2;

    // ---- GEMM1 + exact gelu: 16x768 output, 48 tiles / 16 waves = 3 each ----
    for (int tt = 0; tt < 3; ++tt) {
        const int n = (wave * 3 + tt) * 16 + laneM;
        v8f acc = {0.f,0.f,0.f,0.f,0.f,0.f,0.f,0.f};
        for (int k0 = 0; k0 < DIM; k0 += 4) {
            const int k = k0 + kOff;
            const v2f a = *(const v2f*)&lnX[laneM * LNPAD + k];
            v2f b;
            b.x = W1[(size_t)k * HID + n];
            b.y = W1[(size_t)(k + 1) * HID + n];
            acc = __builtin_amdgcn_wmma_f32_16x16x4_f32(false, a, false, b,
                                                        (short)0, acc, false, false);
        }
#pragma unroll
        for (int r = 0; r < 8; ++r) {
            const int row = r + half * 8;
            const float u = acc[r];
            U[row * UPAD + n] = 0.5f * u * (1.f + erff(u * 0.70710678118654752f));
        }
    }
    __syncthreads();

    // ---- GEMM2 + residual: 16x192 output, waves 0..11 ----
    if (wave < 12) {
        const int n = wave * 16 + laneM;
        v8f acc = {0.f,0.f,0.f,0.f,0.f,0.f,0.f,0.f};
        for (int k0 = 0; k0 < HID; k0 += 4) {
            const int k = k0 + kOff;
            const v2f a = *(const v2f*)&U[laneM * UPAD + k];
            v2f b;
            b.x = W2[(size_t)k * DIM + n];
            b.y = W2[(size_t)(k + 1) * DIM + n];
            acc = __builtin_amdgcn_wmma_f32_16x16x4_f32(false, a, false, b,
                                                        (short)0, acc, false, false);
        }
#pragma unroll
        for (int r = 0; r < 8; ++r) {
            const int row = r + half * 8;
            const size_t gi = (size_t)(m0 + row) * DIM + n;
            out[gi] = acc[r] + h[gi] + msg[gi];
        }
    }
}

// ---------------------------------------------------------------------------
extern "C" void kernel_launch(void* const* d_in, const int* in_sizes, int n_in,
                              void* d_out, int out_size, void* d_ws, size_t ws_size,
                              hipStream_t stream)
{
    const float* hA   = (const float*)d_in[0];
    const float* hB   = (const float*)d_in[1];
    const int*   src1 = (const int*)d_in[2];
    const int*   dst1 = (const int*)d_in[3];
    const int*   src2 = (const int*)d_in[4];
    const int*   dst2 = (const int*)d_in[5];
    const int*   src3 = (const int*)d_in[6];
    const int*   dst3 = (const int*)d_in[7];
    const float* ew1  = (const float*)d_in[8];
    const float* ew2  = (const float*)d_in[9];
    const float* ew3  = (const float*)d_in[10];
    const float* Wq   = (const float*)d_in[11];
    const float* Wk   = (const float*)d_in[12];
    const float* Wv   = (const float*)d_in[13];
    const float* R1   = (const float*)d_in[14];
    const float* R2   = (const float*)d_in[15];
    const float* R3   = (const float*)d_in[16];
    const float* ln_g = (const float*)d_in[17];
    const float* ln_b = (const float*)d_in[18];
    const float* W1   = (const float*)d_in[19];
    const float* W2   = (const float*)d_in[20];

    const int NA = in_sizes[0] / DIM;
    const int NB = in_sizes[1] / DIM;
    const int E  = in_sizes[2];

    const size_t nA = (size_t)NA * DIM;
    const size_t nB = (size_t)NB * DIM;

    float* ws   = (float*)d_ws;
    float* qA   = ws;          float* kA = qA + nA;  float* vA = kA + nA;
    float* qB   = vA + nA;     float* kB = qB + nB;  float* vB = kB + nB;
    float* msgA = vB + nB;     float* msgB = msgA + nA;

    // zero the scatter-add accumulators (graph-capturable async memset)
    hipMemsetAsync(msgA, 0, (nA + nB) * sizeof(float), stream);

    // node projections: 6 x [N,192]@[192,192] fp32 WMMA GEMMs
    gemm_x_w<<<NA / 16, 384, 0, stream>>>(hA, Wq, qA, NA);
    gemm_x_w<<<NA / 16, 384, 0, stream>>>(hA, Wk, kA, NA);
    gemm_x_w<<<NA / 16, 384, 0, stream>>>(hA, Wv, vA, NA);
    gemm_x_w<<<NB / 16, 384, 0, stream>>>(hB, Wq, qB, NB);
    gemm_x_w<<<NB / 16, 384, 0, stream>>>(hB, Wk, kB, NB);
    gemm_x_w<<<NB / 16, 384, 0, stream>>>(hB, Wv, vB, NB);

    // message passing: rel1 A->B, rel2 B->A, rel3 A->A (accumulates into msgA)
    const int eblocks = (E + 7) / 8;
    edge_msg<<<eblocks, 256, 0, stream>>>(qB, kA, vA, src1, dst1, ew1, R1, msgB, E);
    edge_msg<<<eblocks, 256, 0, stream>>>(qA, kB, vB, src2, dst2, ew2, R2, msgA, E);
    edge_msg<<<eblocks, 256, 0, stream>>>(qA, kA, vA, src3, dst3, ew3, R3, msgA, E);

    // fused residual + LN + FFN
    float* outA = (float*)d_out;
    float* outB = outA + nA;
    ffn_fused<<<NA / 16, 512, 0, stream>>>(hA, msgA, ln_g, ln_b, W1, W2, outA, NA);
    ffn_fused<<<NB / 16, 512, 0, stream>>>(hB, msgB, ln_g, ln_b, W1, W2, outB, NB);
}